// LocalSimilar_20392504721876
// MI455X (gfx1250) — compile-verified
//
#include <hip/hip_runtime.h>
#include <math.h>

typedef float v2f __attribute__((ext_vector_type(2)));
typedef float v8f __attribute__((ext_vector_type(8)));

#define STRD 132   // LDS row stride in floats (132 % 64 == 4 -> conflict-free frags)
#define TP   112   // padded tile positions (100 real: 5 rows x 20 cols)

__global__ __launch_bounds__(256) void
LocalSimilar_fused_kernel(const float* __restrict__ fmap,
                          const float* __restrict__ Wqk,
                          float* __restrict__ out)
{
    __shared__ float fmapT[TP * STRD];   // [pos][ch]  (B operand for projections)
    __shared__ float kT[TP * STRD];      // [pos][ch]  k vectors for halo positions
    __shared__ float qT[16 * STRD];      // [pos][ch]  scaled q for 16 center pixels
    __shared__ float simS[16 * 26];      // [pos][l]   similarity scores

    const int t    = threadIdx.x;
    const int lane = t & 31;
    const int w    = t >> 5;       // wave id 0..7
    const int lo   = lane & 15;
    const int hi   = lane >> 4;

    const int b  = blockIdx.z;         // batch 0..3
    const int x  = blockIdx.y;         // row 0..95
    const int y0 = blockIdx.x * 16;    // column block base

    // ---- Phase 0: load edge-clamped fmap halo (5 rows x 20 cols x 128 ch) ----
    // Each 32-lane row job loads one (c,r): 20 consecutive gy -> coalesced.
    for (int i = t; i < 128 * 5 * 32; i += 256) {
        int lj  = i & 31;
        int row = i >> 5;              // 0..639
        int c = row / 5, r = row % 5;
        if (lj < 20) {
            int gx = x - 2 + r;   gx = gx < 0 ? 0 : (gx > 95 ? 95 : gx);
            int gy = y0 - 2 + lj; gy = gy < 0 ? 0 : (gy > 127 ? 127 : gy);
            fmapT[(r * 20 + lj) * STRD + c] =
                fmap[(((size_t)b * 128 + c) * 96 + gx) * 128 + gy];
        }
    }
    __syncthreads();

    // ---- Phase 1: q projection. Wave w -> out-channels [16w,16w+16), N = 16 centers.
    // Center pixels are tile positions 42..57 (r=2, j=2..17) -> contiguous N.
    {
        float2 af[32];
        const float* arow = Wqk + ((size_t)(w * 16 + lo)) * 128 + 2 * hi;
        #pragma unroll
        for (int kk = 0; kk < 32; ++kk) af[kk] = *(const float2*)(arow + kk * 4);

        v8f acc = {0.f, 0.f, 0.f, 0.f, 0.f, 0.f, 0.f, 0.f};
        const float* brow = fmapT + (42 + lo) * STRD + 2 * hi;
        #pragma unroll
        for (int kk = 0; kk < 32; ++kk) {
            float2 b2 = *(const float2*)(brow + kk * 4);
            v2f av; av.x = af[kk].x; av.y = af[kk].y;
            v2f bv; bv.x = b2.x;     bv.y = b2.y;
            acc = __builtin_amdgcn_wmma_f32_16x16x4_f32(
                false, av, false, bv, (short)0, acc, false, false);
        }
        const float scale = 0.08838834764831845f;   // 128^-0.5
        #pragma unroll
        for (int v = 0; v < 8; ++v)
            qT[lo * STRD + w * 16 + v + 8 * hi] = acc[v] * scale;
    }

    // ---- Phase 2: k projection. Wave w -> k channels [16w,16w+16) for all halo
    // positions (7 N-tiles of 16; positions >=100 are padding, never consumed).
    {
        float2 af[32];
        const float* arow = Wqk + ((size_t)(128 + w * 16 + lo)) * 128 + 2 * hi;
        #pragma unroll
        for (int kk = 0; kk < 32; ++kk) af[kk] = *(const float2*)(arow + kk * 4);

        for (int nt = 0; nt < 7; ++nt) {
            v8f acc = {0.f, 0.f, 0.f, 0.f, 0.f, 0.f, 0.f, 0.f};
            const float* brow = fmapT + (nt * 16 + lo) * STRD + 2 * hi;
            #pragma unroll
            for (int kk = 0; kk < 32; ++kk) {
                float2 b2 = *(const float2*)(brow + kk * 4);
                v2f av; av.x = af[kk].x; av.y = af[kk].y;
                v2f bv; bv.x = b2.x;     bv.y = b2.y;
                acc = __builtin_amdgcn_wmma_f32_16x16x4_f32(
                    false, av, false, bv, (short)0, acc, false, false);
            }
            #pragma unroll
            for (int v = 0; v < 8; ++v)
                kT[(nt * 16 + lo) * STRD + w * 16 + v + 8 * hi] = acc[v];
        }
    }
    __syncthreads();

    // ---- Phase 3: similarity. 16 pixels x 25 neighbors, dot over 128 ch in LDS.
    for (int idx = t; idx < 16 * 25; idx += 256) {
        int pos = idx / 25, l = idx % 25;
        int di = l / 5, dj = l % 5;
        int npos = di * 20 + pos + dj;          // neighbor tile position
        const float4* qv = (const float4*)(qT + pos  * STRD);
        const float4* kv = (const float4*)(kT + npos * STRD);
        float s = 0.f;
        #pragma unroll
        for (int c4 = 0; c4 < 32; ++c4) {
            float4 q4 = qv[c4], k4 = kv[c4];
            s += q4.x * k4.x + q4.y * k4.y + q4.z * k4.z + q4.w * k4.w;
        }
        simS[pos * 26 + l] = s;
    }
    __syncthreads();

    // ---- Phase 4: softmax over 25 neighbors, one thread per pixel.
    if (t < 16) {
        float m = -INFINITY;
        #pragma unroll
        for (int l = 0; l < 25; ++l) m = fmaxf(m, simS[t * 26 + l]);
        float e[25], sum = 0.f;
        #pragma unroll
        for (int l = 0; l < 25; ++l) {
            e[l] = __expf(simS[t * 26 + l] - m);
            sum += e[l];
        }
        float inv = 1.0f / sum;
        float* op = out + ((size_t)b * 12288 + (size_t)(x * 128 + y0 + t)) * 25;
        #pragma unroll
        for (int l = 0; l < 25; ++l) op[l] = e[l] * inv;
    }
}

extern "C" void kernel_launch(void* const* d_in, const int* in_sizes, int n_in,
                              void* d_out, int out_size, void* d_ws, size_t ws_size,
                              hipStream_t stream) {
    (void)in_sizes; (void)n_in; (void)d_ws; (void)ws_size; (void)out_size;
    const float* fmap = (const float*)d_in[0];   // (4,128,96,128) f32
    const float* Wqk  = (const float*)d_in[1];   // (256,128) f32
    float* out = (float*)d_out;                  // (4,1,12288,25) f32

    dim3 grid(128 / 16, 96, 4);   // (y-blocks, rows, batch)
    LocalSimilar_fused_kernel<<<grid, 256, 0, stream>>>(fmap, Wqk, out);
}